// GATv2_37477884624979
// MI455X (gfx1250) — compile-verified
//
#include <hip/hip_runtime.h>

typedef __attribute__((ext_vector_type(2))) float v2f;
typedef __attribute__((ext_vector_type(8))) float v8f;

#define NNODES 100000
#define NEDGES 1600000
#define ET     (NEDGES + NNODES)   // 1,700,000 (with self loops)

__device__ __forceinline__ float leaky(float x) { return x > 0.f ? x : 0.2f * x; }

// Order-preserving float->uint encoding for atomicMax-based segment max.
__device__ __forceinline__ unsigned fenc(float f) {
    unsigned u = __float_as_uint(f);
    return (u & 0x80000000u) ? ~u : (u | 0x80000000u);
}
__device__ __forceinline__ float fdec(unsigned k) {
    unsigned u = (k & 0x80000000u) ? (k & 0x7fffffffu) : ~k;
    return __uint_as_float(u);
}

// ---------------------------------------------------------------------------
// Per-layer init: zero the aggregation accumulator, softmax denominators, and
// set encoded segment-max to "-inf" (encoded 0).
// ---------------------------------------------------------------------------
__global__ void init_layer(float* __restrict__ acc, float* __restrict__ denom,
                           unsigned* __restrict__ emax) {
    int i = blockIdx.x * blockDim.x + threadIdx.x;
    if (i < NNODES * 128) acc[i] = 0.f;
    if (i < NNODES * 2) { denom[i] = 0.f; emax[i] = 0u; }
}

// ---------------------------------------------------------------------------
// WMMA f32 GEMM: Y[M,NC] = act(X[M,KD] @ W[KD,NC] + bias), M % 16 == 0.
// One block = 16 rows; wave w handles output cols [16w,16w+16).
// A tile staged in LDS with stride KD+1 (conflict-free across 64 banks).
// Fragment layouts per CDNA5 ISA 7.12.2 (16x16x4 f32).
// ---------------------------------------------------------------------------
template <int KD, int NC, bool RELU>
__global__ void wmma_gemm(const float* __restrict__ X, const float* __restrict__ W,
                          const float* __restrict__ bias, float* __restrict__ Y) {
    __shared__ float As[16 * (KD + 1)];
    const int tid = threadIdx.x;
    const int rowBase = blockIdx.x * 16;

    // Cooperative load of the 16 x KD activation tile.
    for (int idx = tid; idx < 16 * KD; idx += blockDim.x) {
        int r = idx / KD, c = idx % KD;
        As[r * (KD + 1) + c] = X[(size_t)(rowBase + r) * KD + c];
    }
    __syncthreads();

    const int wave = tid >> 5;
    const int lane = tid & 31;
    const int half = lane >> 4;   // 0: K even pair / rows 0-7 ; 1: K odd pair / rows 8-15
    const int r    = lane & 15;   // A row within tile, also output column within tile
    const int j    = wave * 16 + r;  // absolute output column

    v8f acc;
    const float bv = bias ? bias[j] : 0.f;
#pragma unroll
    for (int v = 0; v < 8; ++v) acc[v] = bv;

#pragma unroll
    for (int k = 0; k < KD; k += 4) {
        v2f a, b;
        // A frag: a[v] = A[r][k + 2*half + v]
        a[0] = As[r * (KD + 1) + k + 2 * half + 0];
        a[1] = As[r * (KD + 1) + k + 2 * half + 1];
        // B frag: b[v] = W[k + 2*half + v][j]
        b[0] = W[(size_t)(k + 2 * half + 0) * NC + j];
        b[1] = W[(size_t)(k + 2 * half + 1) * NC + j];
        acc = __builtin_amdgcn_wmma_f32_16x16x4_f32(false, a, false, b,
                                                    (short)0, acc, false, false);
    }

#pragma unroll
    for (int v = 0; v < 8; ++v) {
        float val = acc[v];
        if (RELU) val = fmaxf(val, 0.f);
        // D frag: d[v] = D[v + 8*half][j]
        Y[(size_t)(rowBase + v + 8 * half) * NC + j] = val;
    }
}

// ---------------------------------------------------------------------------
// Pass 1: attention logits e[eid][h] = att_h . leaky_relu(xl[src] + xr[dst]),
// plus atomic segment max per (dst, head). One wave32 per edge; lane covers
// 4 channels; lanes 0-15 = head 0, lanes 16-31 = head 1.
// ---------------------------------------------------------------------------
__global__ void edge_scores(const int* __restrict__ ei, const float* __restrict__ xl,
                            const float* __restrict__ xr, const float* __restrict__ att_l,
                            float* __restrict__ esc, unsigned* __restrict__ emax) {
    int eid = (blockIdx.x * blockDim.x + threadIdx.x) >> 5;
    if (eid >= ET) return;
    const int lane = threadIdx.x & 31;
    int src, dst;
    if (eid < NEDGES) { src = ei[eid]; dst = ei[NEDGES + eid]; }
    else              { src = dst = eid - NEDGES; }

    const int c4 = lane * 4;
    const float4 lv = *(const float4*)(xl + (size_t)src * 128 + c4);
    const float4 rv = *(const float4*)(xr + (size_t)dst * 128 + c4);
    const float4 av = *(const float4*)(att_l + c4);

    float s = av.x * leaky(lv.x + rv.x) + av.y * leaky(lv.y + rv.y)
            + av.z * leaky(lv.z + rv.z) + av.w * leaky(lv.w + rv.w);

    // Reduce within each 16-lane half (one head each) — wave32 xor shuffles.
    s += __shfl_xor(s, 1, 32);
    s += __shfl_xor(s, 2, 32);
    s += __shfl_xor(s, 4, 32);
    s += __shfl_xor(s, 8, 32);

    if ((lane & 15) == 0) {
        const int h = lane >> 4;
        esc[(size_t)eid * 2 + h] = s;
        atomicMax(&emax[dst * 2 + h], fenc(s));
    }
}

// ---------------------------------------------------------------------------
// Pass 2: ex = exp(e - max[dst]) in-place; accumulate softmax denominator.
// One thread per (edge, head).
// ---------------------------------------------------------------------------
__global__ void edge_exp(const int* __restrict__ ei, float* __restrict__ esc,
                         const unsigned* __restrict__ emax, float* __restrict__ denom) {
    long idx = (long)blockIdx.x * blockDim.x + threadIdx.x;
    if (idx >= (long)ET * 2) return;
    const int eid = (int)(idx >> 1);
    const int h   = (int)(idx & 1);
    const int dst = (eid < NEDGES) ? ei[NEDGES + eid] : (eid - NEDGES);
    const float m = fdec(emax[dst * 2 + h]);
    const float x = __expf(esc[idx] - m);
    esc[idx] = x;
    unsafeAtomicAdd(&denom[dst * 2 + h], x);
}

// ---------------------------------------------------------------------------
// Pass 3: acc[dst] += (ex/denom[dst]) * xl[src]. One wave per edge, 4 channels
// per lane, native f32 global atomics (L2-resident targets).
// ---------------------------------------------------------------------------
__global__ void edge_aggregate(const int* __restrict__ ei, const float* __restrict__ esc,
                               const float* __restrict__ denom, const float* __restrict__ xl,
                               float* __restrict__ acc) {
    int eid = (blockIdx.x * blockDim.x + threadIdx.x) >> 5;
    if (eid >= ET) return;
    const int lane = threadIdx.x & 31;
    int src, dst;
    if (eid < NEDGES) { src = ei[eid]; dst = ei[NEDGES + eid]; }
    else              { src = dst = eid - NEDGES; }

    const int h = lane >> 4;
    const float alpha = esc[(size_t)eid * 2 + h] / denom[dst * 2 + h];
    const int c4 = lane * 4;
    const float4 lv = *(const float4*)(xl + (size_t)src * 128 + c4);
    float* o = acc + (size_t)dst * 128 + c4;
    unsafeAtomicAdd(o + 0, alpha * lv.x);
    unsafeAtomicAdd(o + 1, alpha * lv.y);
    unsafeAtomicAdd(o + 2, alpha * lv.z);
    unsafeAtomicAdd(o + 3, alpha * lv.w);
}

// Concat layers: h = relu(acc + bias12[l])
__global__ void finalize_concat(const float* __restrict__ acc, const float* __restrict__ bias,
                                float* __restrict__ h) {
    int i = blockIdx.x * blockDim.x + threadIdx.x;
    if (i >= NNODES * 128) return;
    h[i] = fmaxf(acc[i] + bias[i & 127], 0.f);
}

// Mean layer: h = relu(mean over heads + bias3)  (second relu is identity)
__global__ void finalize_mean(const float* __restrict__ acc, const float* __restrict__ bias3,
                              float* __restrict__ h) {
    int i = blockIdx.x * blockDim.x + threadIdx.x;
    if (i >= NNODES * 64) return;
    const int n = i >> 6, c = i & 63;
    const float v = 0.5f * (acc[(size_t)n * 128 + c] + acc[(size_t)n * 128 + 64 + c]) + bias3[c];
    h[i] = fmaxf(v, 0.f);
}

extern "C" void kernel_launch(void* const* d_in, const int* in_sizes, int n_in,
                              void* d_out, int out_size, void* d_ws, size_t ws_size,
                              hipStream_t stream) {
    const float* x      = (const float*)d_in[0];
    const float* Wl     = (const float*)d_in[1];
    const float* bl     = (const float*)d_in[2];
    const float* Wr     = (const float*)d_in[3];
    const float* br     = (const float*)d_in[4];
    const float* att    = (const float*)d_in[5];
    const float* bias12 = (const float*)d_in[6];
    const float* bias3  = (const float*)d_in[7];
    const float* lin1_w = (const float*)d_in[8];
    const float* lin1_b = (const float*)d_in[9];
    const float* lin2_w = (const float*)d_in[10];
    const float* lin2_b = (const float*)d_in[11];
    const int*   ei     = (const int*)d_in[12];

    // Workspace carve-up (floats)
    float* xl    = (float*)d_ws;            // N*128
    float* xr    = xl + (size_t)NNODES * 128;
    float* hbuf  = xr + (size_t)NNODES * 128;
    float* acc   = hbuf + (size_t)NNODES * 128;
    float* esc   = acc + (size_t)NNODES * 128;     // ET*2
    float* denom = esc + (size_t)ET * 2;           // N*2
    unsigned* emax = (unsigned*)(denom + (size_t)NNODES * 2);  // N*2
    float* hm = xl;  // N*64, reuse after layer 3
    float* y1 = xr;  // N*64, reuse

    const int MT = NNODES / 16;                 // 6250 row tiles (exact)
    const dim3 b256(256), b128(128);
    const int gInit = (NNODES * 128 + 255) / 256;          // covers acc + denom/emax
    const int gWave = (ET * 32) / 256;                     // 212500 exact
    const int gEH   = (int)(((long)ET * 2 + 255) / 256);
    const int gN128 = (NNODES * 128) / 256;
    const int gN64  = (NNODES * 64) / 256;

    const float* hin = x;
    for (int l = 0; l < 3; ++l) {
        const float* Wl_l = Wl + (size_t)l * 128 * 128;
        const float* Wr_l = Wr + (size_t)l * 128 * 128;
        const float* bl_l = bl + (size_t)l * 128;
        const float* br_l = br + (size_t)l * 128;
        const float* att_l = att + (size_t)l * 128;

        init_layer<<<gInit, b256, 0, stream>>>(acc, denom, emax);

        wmma_gemm<128, 128, false><<<MT, b256, 0, stream>>>(hin, Wl_l, bl_l, xl);
        wmma_gemm<128, 128, false><<<MT, b256, 0, stream>>>(hin, Wr_l, br_l, xr);

        edge_scores<<<gWave, b256, 0, stream>>>(ei, xl, xr, att_l, esc, emax);
        edge_exp<<<gEH, b256, 0, stream>>>(ei, esc, emax, denom);
        edge_aggregate<<<gWave, b256, 0, stream>>>(ei, esc, denom, xl, acc);

        if (l < 2) {
            finalize_concat<<<gN128, b256, 0, stream>>>(acc, bias12 + (size_t)l * 128, hbuf);
            hin = hbuf;
        } else {
            finalize_mean<<<gN64, b256, 0, stream>>>(acc, bias3, hm);
        }
    }

    // Final MLP: relu(hm @ lin1 + b1) @ lin2 + b2
    wmma_gemm<64, 64, true ><<<MT, b128, 0, stream>>>(hm, lin1_w, lin1_b, y1);
    wmma_gemm<64, 64, false><<<MT, b128, 0, stream>>>(y1, lin2_w, lin2_b, (float*)d_out);
}